// ROI_Pool_163208757586
// MI455X (gfx1250) — compile-verified
//
#include <hip/hip_runtime.h>
#include <cstdint>
#include <math.h>

// ROI max-pool, CDNA5 (gfx1250).
// Memory-bound gather: feature map (4 MB) is L2-resident; DRAM floor is
// ~4 MB read + 32 MB write (~1.5 us at 23.3 TB/s). Each block owns one ROI
// and 16 channels. The <=40x40 fp32 ROI window is DMA'd global->LDS by the
// Tensor Data Mover (one TENSOR_LOAD_TO_LDS per channel, TENSORcnt-tracked),
// double-buffered so the DMA of channel i+1 overlaps the bin reduction of
// channel i. All 8 wave32 waves then reduce 8x8 bins entirely out of LDS.

#define C_DIM        256
#define H_DIM        64
#define W_DIM        64
#define PH_BINS      8
#define PW_BINS      8
#define WMAX         40          // max window extent (ro <= 20 -> L <= 40)
#define CH_PER_BLOCK 16
#define NTHREADS     256

typedef __attribute__((address_space(3))) void* as3_ptr_t;
typedef unsigned int u32x4 __attribute__((ext_vector_type(4)));
typedef int          i32x4 __attribute__((ext_vector_type(4)));
typedef int          i32x8 __attribute__((ext_vector_type(8)));

#if __has_builtin(__builtin_amdgcn_s_wait_tensorcnt)
#define WAIT_TENSOR(n) __builtin_amdgcn_s_wait_tensorcnt(n)
#else
#define WAIT_TENSOR(n) asm volatile("s_wait_tensorcnt %0" ::"i"(n) : "memory")
#endif

// One TDM 2D tile load: Lx x Ly fp32 tile at gptr (row stride W_DIM dwords)
// -> packed row-major tile at LDS byte address lds_addr. D# built per the
// CDNA5 ISA bit layout (groups 2/3 zero => 2D tensor, tile_dim2=0 unused).
__device__ __forceinline__ void tdm_load_window(const float* gptr,
                                                unsigned lds_addr,
                                                int Lx, int Ly) {
  const unsigned long long ga = (unsigned long long)(uintptr_t)gptr;

  u32x4 g0;
  g0.x = 1u;                                    // [1:0] count=1 (valid user D#)
  g0.y = lds_addr;                              // [63:32] lds_addr (bytes)
  g0.z = (unsigned)(ga & 0xFFFFFFFFu);          // [95:64] global_addr lo
  g0.w = ((unsigned)(ga >> 32) & 0x01FFFFFFu)   // [120:96] global_addr hi
         | (2u << 30);                          // [127:126] type=2 ("image")

  i32x8 g1;
  g1[0] = 0x00020000;                           // data_size=2 (4B); mask/pad/iter=0
  g1[1] = (Lx & 0xFFFF) << 16;                  // [63:48] tensor_dim0 lo16
  g1[2] = (((unsigned)Lx >> 16) & 0xFFFF)       // [79:64] tensor_dim0 hi16
          | ((Ly & 0xFFFF) << 16);              // [95:80] tensor_dim1 lo16
  g1[3] = (((unsigned)Ly >> 16) & 0xFFFF)       // [111:96] tensor_dim1 hi16
          | ((Lx & 0xFFFF) << 16);              // [127:112] tile_dim0 = Lx
  g1[4] = (Ly & 0xFFFF);                        // [143:128] tile_dim1 = Ly; tile_dim2=0
  g1[5] = W_DIM;                                // [191:160] tensor_dim0_stride lo32
  g1[6] = 0;                                    // stride0 hi16 / stride1 lo16
  g1[7] = 0;                                    // stride1 hi32

  const i32x4 z4 = {0, 0, 0, 0};
#if __clang_major__ >= 23
  const i32x8 z8 = {0, 0, 0, 0, 0, 0, 0, 0};    // amdgpu-toolchain 6-arg form
  __builtin_amdgcn_tensor_load_to_lds(g0, g1, z4, z4, z8, 0);
#else
  __builtin_amdgcn_tensor_load_to_lds(g0, g1, z4, z4, 0);  // ROCm 7.2 5-arg form
#endif
}

__global__ __launch_bounds__(NTHREADS) void roi_pool_kernel(
    const float* __restrict__ fm, const int* __restrict__ rois,
    float* __restrict__ out, int n_rois) {
  __shared__ float win[2][WMAX * WMAX];
  __shared__ float part[NTHREADS];

  const int n   = blockIdx.x;
  const int c0  = blockIdx.y * CH_PER_BLOCK;
  const int tid = threadIdx.x;

  // ROI geometry (uniform per block -> SALU).
  const int ry = rois[4 * n + 0];
  const int rx = rois[4 * n + 1];
  const int rh = rois[4 * n + 2];
  const int rw = rois[4 * n + 3];
  const int roh = rh >> 1, row = rw >> 1;
  int ymin = ry - roh; if (ymin < 0) ymin = 0;
  int ymax = ry + roh; if (ymax > H_DIM) ymax = H_DIM - 1;
  int xmin = rx - row; if (xmin < 0) xmin = 0;
  int xmax = rx + row; if (xmax > W_DIM) xmax = W_DIM - 1;
  const int Ly = ymax - ymin;
  const int Lx = xmax - xmin;   // window always fully inside the feature map

  // Per-thread bin geometry: 4 threads per bin split the bin's rows.
  const int bin = tid & 63;
  const int prt = tid >> 6;
  const int ph  = bin >> 3, pw = bin & 7;
  const int rs  = (ph * Ly) >> 3;             // window-relative row start
  const int re  = ((ph + 1) * Ly + 7) >> 3;   // window-relative row end
  const int cs  = (pw * Lx) >> 3;             // window-relative col start
  const int ce  = ((pw + 1) * Lx + 7) >> 3;   // window-relative col end
  const int ncols = ce - cs;

  const float* g_win0 = fm + (size_t)c0 * (H_DIM * W_DIM) +
                        (size_t)ymin * W_DIM + xmin;
  const unsigned lds_buf[2] = {
      (unsigned)(uintptr_t)(as3_ptr_t)&win[0][0],
      (unsigned)(uintptr_t)(as3_ptr_t)&win[1][0]};

  const bool issuer = (tid < 32);  // wave 0 only (wave32); uniform per wave

  // Prologue: kick DMA of channel 0's window.
  if (issuer) tdm_load_window(g_win0, lds_buf[0], Lx, Ly);

  for (int i = 0; i < CH_PER_BLOCK; ++i) {
    if (issuer) {
      if (i + 1 < CH_PER_BLOCK) {
        // Kick DMA for next channel, then drain exactly the older tile:
        // tensor ops from one wave complete in order, so TENSORcnt<=1 means
        // channel i's tile has fully landed in LDS.
        tdm_load_window(g_win0 + (size_t)(i + 1) * (H_DIM * W_DIM),
                        lds_buf[(i + 1) & 1], Lx, Ly);
        WAIT_TENSOR(1);
      } else {
        WAIT_TENSOR(0);
      }
    }
    __syncthreads();  // publish wave 0's landed tile to all waves

    // Pass A: partial max over this part's rows of the bin, from LDS.
    // TDM stores the tile packed row-major with row stride Lx.
    const float* w = win[i & 1];
    float m = -INFINITY;
    for (int y = rs + prt; y < re; y += 4) {
      const float* rowp = w + y * Lx + cs;
      for (int x = 0; x < ncols; ++x) m = fmaxf(m, rowp[x]);
    }
    part[tid] = m;
    __syncthreads();

    // Pass B: combine 4 partials per bin; coalesced 64-dword store.
    if (tid < 64) {
      float v = fmaxf(fmaxf(part[tid], part[tid + 64]),
                      fmaxf(part[tid + 128], part[tid + 192]));
      out[((size_t)n * C_DIM + (c0 + i)) * (PH_BINS * PW_BINS) + tid] = v;
    }
    __syncthreads();  // protect part[] and the LDS buffer being re-filled next
  }
  (void)n_rois;
}

extern "C" void kernel_launch(void* const* d_in, const int* in_sizes, int n_in,
                              void* d_out, int out_size, void* d_ws, size_t ws_size,
                              hipStream_t stream) {
  const float* fm  = (const float*)d_in[0];
  const int* rois  = (const int*)d_in[1];
  float* out       = (float*)d_out;
  const int n_rois = in_sizes[1] / 4;                // 512
  const int c_dim  = in_sizes[0] / (H_DIM * W_DIM);  // 256

  dim3 grid(n_rois, c_dim / CH_PER_BLOCK);
  dim3 block(NTHREADS);
  roi_pool_kernel<<<grid, block, 0, stream>>>(fm, rois, out, n_rois);
  (void)n_in; (void)out_size; (void)d_ws; (void)ws_size;
}